// BPR_LOSS_56143812493724
// MI455X (gfx1250) — compile-verified
//
#include <hip/hip_runtime.h>
#include <cmath>

typedef __attribute__((ext_vector_type(2))) float v2f;
typedef __attribute__((ext_vector_type(8))) float v8f;

#define B_DIM 16
#define N_DIM 2048

// Workspace layout:
//   [0,   192): float S[B][3]   level sums
//   [256, 512): int   cnt[B][4] label class counts
// -> zero first 512 bytes (128 dwords) every launch.

__global__ void zero_ws_kernel(unsigned* ws) {
  ws[threadIdx.x] = 0u;  // 128 threads x 4B = 512B
}

__global__ void count_kernel(const int* __restrict__ labels, int* __restrict__ cnt) {
  const int b = blockIdx.x;
  const int t = threadIdx.x;
  int local[4] = {0, 0, 0, 0};
  for (int i = t; i < N_DIM; i += blockDim.x) {
    int l = labels[b * N_DIM + i] & 3;
    local[l]++;
  }
#pragma unroll
  for (int k = 0; k < 4; ++k)
    if (local[k]) atomicAdd(&cnt[b * 4 + k], local[k]);
}

// Raw hardware transcendentals (gfx1250 v_exp_f32 / v_log_f32 = exp2 / log2).
// s_delay_alu TRANS32_DEP_1 satisfies the CDNA5 multicycle-TRANS RAW hazard
// (ISA 7.4: one independent op / delay before the TRANS result is consumed).
__device__ __forceinline__ float hw_exp2(float x) {
  float r;
  asm("v_exp_f32_e32 %0, %1\n\t"
      "s_delay_alu instid0(TRANS32_DEP_1)"
      : "=v"(r) : "v"(x));
  return r;
}
__device__ __forceinline__ float hw_log2(float x) {
  float r;
  asm("v_log_f32_e32 %0, %1\n\t"
      "s_delay_alu instid0(TRANS32_DEP_1)"
      : "=v"(r) : "v"(x));
  return r;
}

// log_sigmoid(x) = min(x,0) - log(1 + exp(-|x|)).
// exp arg <= 0 -> e in (0,1]; log arg in [1,2] -> no denorm/zero guards needed.
__device__ __forceinline__ float fast_logsigmoid(float x) {
  const float NLOG2E = -1.44269504088896340736f;  // -log2(e)
  const float LN2    =  0.69314718055994530942f;  // ln(2)
  float e = hw_exp2(fabsf(x) * NLOG2E);           // exp(-|x|)
  float l = hw_log2(1.0f + e);                    // log2(1+e)
  return fminf(x, 0.0f) - l * LN2;
}

__global__ void __launch_bounds__(256)
pair_kernel(const float* __restrict__ logits, const int* __restrict__ labels,
            float* __restrict__ S) {
  const int b    = blockIdx.x >> 4;   // batch
  const int rg   = blockIdx.x & 15;   // row-tile group (8 tiles per block)
  const int wave = threadIdx.x >> 5;  // 8 waves of 32
  const int lane = threadIdx.x & 31;
  const int hi   = lane >> 4;         // half-wave select
  const int ln   = lane & 15;

  const float* lg = logits + b * N_DIM;
  const int*   lb = labels + b * N_DIM;

  // Stage this batch's logits + labels into LDS (16 KB of 320 KB/WGP).
  __shared__ float sLog[N_DIM];
  __shared__ int   sLab[N_DIM];
  for (int i = threadIdx.x; i < N_DIM; i += 256) {
    sLog[i] = lg[i];
    sLab[i] = lb[i];
  }
  __syncthreads();

  const int rowbase = (rg * 8 + wave) * 16;

  // A-matrix (16x4 f32): row i = (l_i, 1, 0, 0).
  // Layout: lanes 0-15 hold K=0,1 (VGPR0,1) for M=lane; lanes 16-31 hold K=2,3.
  const float lrow = sLog[rowbase + ln];
  v2f Amat;
  Amat.x = hi ? 0.0f : lrow;  // K=0 (upper half K=2 -> 0)
  Amat.y = hi ? 0.0f : 1.0f;  // K=1 (upper half K=3 -> 0)

  // Row labels for this lane's 8 C/D elements: row = rowbase + v + 8*hi.
  int rlab[8];
#pragma unroll
  for (int v = 0; v < 8; ++v) rlab[v] = sLab[rowbase + v + 8 * hi];

  float acc0 = 0.0f, acc1 = 0.0f, acc2 = 0.0f;

  for (int ct = 0; ct < N_DIM / 16; ++ct) {
    const int colbase = ct * 16;
    const float lcol = sLog[colbase + ln];
    const int   clab = sLab[colbase + ln];

    // B-matrix (4x16 f32): col j = (1, -l_j, 0, 0)^T.
    // VGPR0: lanes 0-15 = (K=0,N=ln), lanes 16-31 = (K=2,N=ln); VGPR1 = K=1 / K=3.
    v2f Bmat;
    Bmat.x = hi ? 0.0f : 1.0f;   // K=0
    Bmat.y = hi ? 0.0f : -lcol;  // K=1

    v8f c = {};
    // D[i][j] = l_i*1 + 1*(-l_j) = l_i - l_j  (exact f32 delta tile)
    v8f d = __builtin_amdgcn_wmma_f32_16x16x4_f32(
        /*neg_a=*/false, Amat, /*neg_b=*/false, Bmat,
        /*c_mod=*/(short)0, c, /*reuse_a=*/false, /*reuse_b=*/false);

#pragma unroll
    for (int v = 0; v < 8; ++v) {
      float ls = fast_logsigmoid(d[v]);
      int   a  = rlab[v];
      float m  = (a > clab) ? ls : 0.0f;  // contributes to level a iff label_col < a
      acc0 += (a == 1) ? m : 0.0f;
      acc1 += (a == 2) ? m : 0.0f;
      acc2 += (a == 3) ? m : 0.0f;
    }
  }

  // wave32 all-lane reduction
#pragma unroll
  for (int off = 16; off >= 1; off >>= 1) {
    acc0 += __shfl_xor(acc0, off, 32);
    acc1 += __shfl_xor(acc1, off, 32);
    acc2 += __shfl_xor(acc2, off, 32);
  }
  if (lane == 0) {
    atomicAdd(&S[b * 3 + 0], acc0);
    atomicAdd(&S[b * 3 + 1], acc1);
    atomicAdd(&S[b * 3 + 2], acc2);
  }
}

__global__ void final_kernel(const float* __restrict__ S, const int* __restrict__ cnt,
                             float* __restrict__ out) {
  const int b = threadIdx.x;  // one wave32
  float pg = 0.0f;
  if (b < B_DIM) {
    const int c0 = cnt[b * 4 + 0], c1 = cnt[b * 4 + 1];
    const int c2 = cnt[b * 4 + 2], c3 = cnt[b * 4 + 3];
    const float sums[3] = {S[b * 3 + 0], S[b * 3 + 1], S[b * 3 + 2]};
    const int pos[3] = {c1, c2, c3};
    const int neg[3] = {c0, c0 + c1, c0 + c1 + c2};
    float tot = 0.0f, nv = 0.0f;
#pragma unroll
    for (int k = 0; k < 3; ++k) {
      float pairs = (float)pos[k] * (float)neg[k];
      if (pairs > 0.0f) { tot += sums[k] / pairs; nv += 1.0f; }
    }
    pg = tot / fmaxf(nv, 1.0f);
  }
#pragma unroll
  for (int off = 16; off >= 1; off >>= 1) pg += __shfl_xor(pg, off, 32);
  if (threadIdx.x == 0) out[0] = -(pg / (float)B_DIM);
}

extern "C" void kernel_launch(void* const* d_in, const int* in_sizes, int n_in,
                              void* d_out, int out_size, void* d_ws, size_t ws_size,
                              hipStream_t stream) {
  const float* logits = (const float*)d_in[0];
  const int*   labels = (const int*)d_in[1];
  float* S   = (float*)d_ws;
  int*   cnt = (int*)((char*)d_ws + 256);
  float* out = (float*)d_out;

  zero_ws_kernel<<<1, 128, 0, stream>>>((unsigned*)d_ws);
  count_kernel<<<B_DIM, 256, 0, stream>>>(labels, cnt);
  // 16 blocks per batch x 8 waves/block = 128 row tiles of 16 -> N=2048 rows
  pair_kernel<<<B_DIM * 16, 256, 0, stream>>>(logits, labels, S);
  final_kernel<<<1, 32, 0, stream>>>(S, cnt, out);
}